// AggregationEncoder_12704513262329
// MI455X (gfx1250) — compile-verified
//
#include <hip/hip_runtime.h>
#include <cstdint>
#include <cstddef>

// Problem constants (match reference).
constexpr int B = 4;
constexpr int G = 65160;
constexpr int F = 128;       // floats per row; 512 bytes = 32 lanes x float4
constexpr int M = 10242;
constexpr int E = 262144;    // even, power of two -> edge pairs never straddle a batch
constexpr int WAVES_PER_BLOCK = 8;
constexpr int BLOCK = WAVES_PER_BLOCK * 32;

// ---------------------------------------------------------------- zero init
__global__ void aggenc_zero_kernel(float* __restrict__ out, float* __restrict__ cnt) {
  const long n = (long)B * M * F;
  const long nc = (long)B * M;
  const long stride = (long)gridDim.x * blockDim.x;
  for (long k = (long)blockIdx.x * blockDim.x + threadIdx.x; k < n; k += stride)
    out[k] = 0.0f;
  for (long k = (long)blockIdx.x * blockDim.x + threadIdx.x; k < nc; k += stride)
    cnt[k] = 0.0f;
}

// ------------------------------------------------- gather + atomic scatter
// One wave per edge-PAIR per stage. Each 512-byte grid row is pulled with a
// single wave-wide global_load_async_to_lds_b128 (CDNA5 async-DMA path,
// ASYNCcnt flow control); two rows per stage, double-buffered slots -> up to
// 4 rows in flight per wave. Consumed rows are scattered with non-returning
// f32 L2 atomics.
__global__ void __launch_bounds__(BLOCK)
aggenc_scatter_kernel(const float* __restrict__ grid,
                      const int2* __restrict__ edges,
                      float* __restrict__ out,
                      float* __restrict__ cnt) {
  // [wave][slot][row-in-pair][F]
  __shared__ __align__(16) float buf[WAVES_PER_BLOCK][2][2][F];

  const int lane = threadIdx.x & 31;
  const int wv   = threadIdx.x >> 5;

  constexpr long totalPairs = (long)B * E / 2;
  constexpr int  EPairs     = E / 2;
  const long nwaves = (long)gridDim.x * WAVES_PER_BLOCK;

  // LDS byte addresses of this lane's 16-byte slice for each slot/row.
  // Low 32 bits of the flat address of a __shared__ object are the LDS offset.
  const uint32_t lds00 = (uint32_t)(uintptr_t)&buf[wv][0][0][lane * 4];
  const uint32_t lds01 = (uint32_t)(uintptr_t)&buf[wv][0][1][lane * 4];
  const uint32_t lds10 = (uint32_t)(uintptr_t)&buf[wv][1][0][lane * 4];
  const uint32_t lds11 = (uint32_t)(uintptr_t)&buf[wv][1][1][lane * 4];

  int curB = 0, curD0 = 0, curD1 = 0;   // metadata for rows being consumed
  int nxtB = 0, nxtD0 = 0, nxtD1 = 0;

  auto issue_pair = [&](long p, uint32_t ldsA, uint32_t ldsB,
                        int& mb, int& md0, int& md1) {
    const int b  = (int)(p / EPairs);
    const int e0 = (int)(p - (long)b * EPairs) * 2;
    // Two adjacent edges: one 16-byte index load (src0,dst0,src1,dst1).
    const int4 ed = *reinterpret_cast<const int4*>(&edges[(size_t)b * E + e0]);
    mb = b; md0 = ed.y; md1 = ed.w;
    const unsigned long long g0 =
        (unsigned long long)(grid + ((size_t)b * G + (size_t)ed.x) * F + lane * 4);
    const unsigned long long g1 =
        (unsigned long long)(grid + ((size_t)b * G + (size_t)ed.z) * F + lane * 4);
    // GV mode: per-lane 64-bit global addr -> per-lane LDS addr, 16B/lane.
    asm volatile("global_load_async_to_lds_b128 %0, %1, off"
                 :: "v"(ldsA), "v"(g0) : "memory");
    asm volatile("global_load_async_to_lds_b128 %0, %1, off"
                 :: "v"(ldsB), "v"(g1) : "memory");
  };

  long i = (long)blockIdx.x * WAVES_PER_BLOCK + wv;
  if (i >= totalPairs) return;

  int slot = 0;
  issue_pair(i, lds00, lds01, curB, curD0, curD1);
  long nxt = i + nwaves;

  while (true) {
    const bool has_next = (nxt < totalPairs);
    if (has_next) {
      issue_pair(nxt, slot ? lds00 : lds10, slot ? lds01 : lds11,
                 nxtB, nxtD0, nxtD1);
      // 2 loads just issued may remain outstanding; the 2 older ones (the
      // pair we are about to consume) must have landed in LDS.
      asm volatile("s_wait_asynccnt 0x2" ::: "memory");
    } else {
      asm volatile("s_wait_asynccnt 0x0" ::: "memory");
    }

    // Consume both rows of the current slot (ds_load_b128) and scatter with
    // non-returning global_atomic_add_f32 (resolved at the L2 atomic units).
    const float4 v0 = *reinterpret_cast<const float4*>(&buf[wv][slot][0][lane * 4]);
    const float4 v1 = *reinterpret_cast<const float4*>(&buf[wv][slot][1][lane * 4]);
    float* o0 = out + ((size_t)curB * M + (size_t)curD0) * F + lane * 4;
    float* o1 = out + ((size_t)curB * M + (size_t)curD1) * F + lane * 4;
    unsafeAtomicAdd(o0 + 0, v0.x);
    unsafeAtomicAdd(o0 + 1, v0.y);
    unsafeAtomicAdd(o0 + 2, v0.z);
    unsafeAtomicAdd(o0 + 3, v0.w);
    unsafeAtomicAdd(o1 + 0, v1.x);
    unsafeAtomicAdd(o1 + 1, v1.y);
    unsafeAtomicAdd(o1 + 2, v1.z);
    unsafeAtomicAdd(o1 + 3, v1.w);
    if (lane == 0) {
      unsafeAtomicAdd(&cnt[(size_t)curB * M + (size_t)curD0], 1.0f);
      unsafeAtomicAdd(&cnt[(size_t)curB * M + (size_t)curD1], 1.0f);
    }

    if (!has_next) break;
    curB = nxtB; curD0 = nxtD0; curD1 = nxtD1;
    slot ^= 1;
    nxt += nwaves;
  }
}

// ---------------------------------------------------------------- mean
__global__ void aggenc_finalize_kernel(float* __restrict__ out,
                                       const float* __restrict__ cnt) {
  const long n = (long)B * M * F;
  const long stride = (long)gridDim.x * blockDim.x;
  for (long k = (long)blockIdx.x * blockDim.x + threadIdx.x; k < n; k += stride) {
    const float c = cnt[k >> 7];          // F == 128
    out[k] = out[k] / fmaxf(c, 1.0f);
  }
}

extern "C" void kernel_launch(void* const* d_in, const int* in_sizes, int n_in,
                              void* d_out, int out_size, void* d_ws, size_t ws_size,
                              hipStream_t stream) {
  (void)in_sizes; (void)n_in; (void)out_size; (void)ws_size;

  const float* grid  = (const float*)d_in[0];
  const int2*  edges = (const int2*)d_in[1];   // [B, E, 2] int32 pairs
  float*       out   = (float*)d_out;          // [B, M, F] accumulators
  float*       cnt   = (float*)d_ws;           // [B, M] edge counts (164 KB)

  aggenc_zero_kernel<<<512, 256, 0, stream>>>(out, cnt);
  aggenc_scatter_kernel<<<2048, BLOCK, 0, stream>>>(grid, edges, out, cnt);
  aggenc_finalize_kernel<<<512, 256, 0, stream>>>(out, cnt);
}